// SpectralMapDecomposition_76063870812684
// MI455X (gfx1250) — compile-verified
//
#include <hip/hip_runtime.h>
#include <math.h>

typedef __attribute__((ext_vector_type(2))) float v2f;
typedef __attribute__((ext_vector_type(8))) float v8f;

#define TWO_PI_OVER_256 (6.2831853071795864769f / 256.0f)

// ---------------------------------------------------------------------------
// Phase 1: Re/Im DFT coefficients via f32 WMMA GEMM.
//   A = img as (M=2048, K=256)  (M = b*256 + h)
//   Bc[k,n] = cos(2*pi*k*n/256), Bs[k,n] = sin(2*pi*k*n/256)  (N = i)
//   Output stored transposed + interleaved: riT[(b*256+i)*256 + h] = {Re, Im}
// ---------------------------------------------------------------------------
__global__ __launch_bounds__(256) void dft_wmma_kernel(const float* __restrict__ img,
                                                       float2* __restrict__ riT) {
    const int lane = threadIdx.x & 31;
    const int wave = threadIdx.x >> 5;
    const int tile = blockIdx.x * 8 + wave;   // 0..2047
    const int m0 = (tile >> 4) << 4;          // M tile base (128 tiles)
    const int n0 = (tile & 15) << 4;          // N tile base (16 tiles)
    const int lo = lane & 15;
    const int hi = lane >> 4;
    const int n  = n0 + lo;

    v8f accC = {};
    v8f accS = {};

    for (int k0 = 0; k0 < 256; k0 += 4) {
        const int kk = k0 + 2 * hi;           // lane-half selects K+0/K+2 pair
        // A operand: a.x = A[m, kk], a.y = A[m, kk+1]
        v2f a = *(const v2f*)(img + (m0 + lo) * 256 + kk);
        // B operands: basis values, exact integer phase reduction mod 256
        const int t0 = (kk * n) & 255;
        const int t1 = ((kk + 1) * n) & 255;
        float s0, c0, s1, c1;
        __sincosf((float)t0 * TWO_PI_OVER_256, &s0, &c0);
        __sincosf((float)t1 * TWO_PI_OVER_256, &s1, &c1);
        v2f bc; bc.x = c0; bc.y = c1;
        v2f bs; bs.x = s0; bs.y = s1;
        accC = __builtin_amdgcn_wmma_f32_16x16x4_f32(false, a, false, bc, (short)0,
                                                     accC, false, false);
        accS = __builtin_amdgcn_wmma_f32_16x16x4_f32(false, a, false, bs, (short)0,
                                                     accS, false, false);
    }

    // D layout: VGPR r, lanes 0-15 -> M = m0+r, lanes 16-31 -> M = m0+r+8; N = n0+(lane&15)
#pragma unroll
    for (int r = 0; r < 8; ++r) {
        const int M    = m0 + r + 8 * hi;
        const int bidx = M >> 8;
        const int h    = M & 255;
        const int i    = n0 + lo;
        float2 v; v.x = accC[r]; v.y = accS[r];
        riT[(bidx << 16) | (i << 8) | h] = v;
    }
}

// ---------------------------------------------------------------------------
// Phase 2: streaming expansion (512 MB). One wave = one 16x16 (h,w) tile for a
// fixed (b,i). Single rank-2 WMMA per tile:
//   A[h, 0..3] = {Re, Im, 0, 0},  B[0..3, w] = {cos/W, sin/W, 0, 0}
// ---------------------------------------------------------------------------
__global__ __launch_bounds__(256) void expand_wmma_kernel(const float2* __restrict__ riT,
                                                          float* __restrict__ out) {
    const int lane  = threadIdx.x & 31;
    const int wave  = threadIdx.x >> 5;
    const int plane = blockIdx.x >> 5;        // b*256 + i
    const int tg    = blockIdx.x & 31;
    const int tile  = tg * 8 + wave;          // 0..255 within plane
    const int h0    = (tile >> 4) << 4;
    const int w0    = (tile & 15) << 4;
    const int bidx  = plane >> 8;
    const int i     = plane & 255;
    const int lo    = lane & 15;
    const int hi    = lane >> 4;

    // A operand: lanes 0-15 carry {Re, Im} for h = h0+lo (contiguous 128B load);
    // lanes 16-31 are the K=2,3 slots -> zero via cndmask (EXEC stays all-1s).
    const float2 ri = riT[(plane << 8) | (h0 + lo)];
    v2f a; a.x = hi ? 0.0f : ri.x;
    a.y = hi ? 0.0f : ri.y;

    // B operand: lanes 0-15 carry {cos, sin}(2*pi*i*w/256)/256 for w = w0+lo.
    const int t = (i * (w0 + lo)) & 255;
    float s, c;
    __sincosf((float)t * TWO_PI_OVER_256, &s, &c);
    const float inv = 1.0f / 256.0f;
    v2f b; b.x = hi ? 0.0f : c * inv;
    b.y = hi ? 0.0f : s * inv;

    v8f acc = {};
    acc = __builtin_amdgcn_wmma_f32_16x16x4_f32(false, a, false, b, (short)0,
                                                acc, false, false);

    // out[b][i][h][w], C = 288 channels total
    const size_t base = ((size_t)bidx * 288 + (size_t)i) * 65536 + (size_t)w0 + lo;
#pragma unroll
    for (int r = 0; r < 8; ++r) {
        out[base + (size_t)(h0 + r + 8 * hi) * 256] = acc[r];
    }
}

// ---------------------------------------------------------------------------
// Phase 3: mask_embedding -> channels [256, 288). float4 streaming copy.
// ---------------------------------------------------------------------------
__global__ __launch_bounds__(256) void mask_copy_kernel(const float4* __restrict__ mask,
                                                        float* __restrict__ out) {
    const size_t f = (size_t)blockIdx.x * 256 + threadIdx.x; // float4 index
    const size_t e = f * 4;                                  // element index
    const size_t per_b = (size_t)32 * 65536;                 // elems per batch in mask
    const size_t b   = e / per_b;
    const size_t rem = e - b * per_b;
    const float4 v = mask[f];
    *(float4*)(out + b * ((size_t)288 * 65536) + (size_t)256 * 65536 + rem) = v;
}

extern "C" void kernel_launch(void* const* d_in, const int* in_sizes, int n_in,
                              void* d_out, int out_size, void* d_ws, size_t ws_size,
                              hipStream_t stream) {
    const float* img  = (const float*)d_in[0];  // (8,1,256,256) f32
    const float* mask = (const float*)d_in[1];  // (8,32,256,256) f32
    float* out = (float*)d_out;                 // (8,288,256,256) f32
    float2* riT = (float2*)d_ws;                // 2048*256 float2 = 4 MB scratch

    // Phase 1: 2048 WMMA tiles, 8 waves/block
    dft_wmma_kernel<<<256, 256, 0, stream>>>(img, riT);
    // Phase 2: 8*256 planes * 256 tiles, 8 waves/block
    expand_wmma_kernel<<<65536, 256, 0, stream>>>(riT, out);
    // Phase 3: 16,777,216 floats as float4
    mask_copy_kernel<<<16384, 256, 0, stream>>>((const float4*)mask, out);
}